// MultiScaleDeformableAttention_11733850652690
// MI455X (gfx1250) — compile-verified
//
#include <hip/hip_runtime.h>
#include <hip/hip_bf16.h>

typedef __attribute__((ext_vector_type(2))) float v2f;
typedef __attribute__((ext_vector_type(8))) float v8f;

#define EMBED  256
#define HEADS  8
#define LEVELS 4
#define POINTS 4
#define DIM    32
#define NQ     13294
#define BS     2
#define MTOT   (BS * NQ)       // 26588 rows

// ---------------- WMMA f32 GEMM: C = A(MxK=256) * B(256xN) + bias (+residual) ----
// Block tile 128(M) x 64(N), K-step 32. 8 waves/block, each wave a 32x32 tile
// built from four 16x16 f32 accumulators via V_WMMA_F32_16X16X4_F32.
#define TM 128
#define TN 64
#define TK 32
#define LDA_S 36   // padded k-stride for As (even -> 8B-aligned v2f loads)
#define LDB_S 36   // padded k-stride for Bs[n][k]

__global__ __launch_bounds__(256)
void msda_gemm_wmma_f32(const float* __restrict__ A, const float* __restrict__ B,
                        const float* __restrict__ bias, const float* __restrict__ residual,
                        float* __restrict__ C, int M, int N)
{
    __shared__ __align__(16) float As[TM * LDA_S];
    __shared__ __align__(16) float Bs[TN * LDB_S];

    const int tid  = threadIdx.x;
    const int lane = tid & 31;
    const int wave = tid >> 5;
    const int wm   = (wave & 3) * 32;   // wave M offset in tile (4 waves along M)
    const int wn   = (wave >> 2) * 32;  // wave N offset in tile (2 waves along N)

    const int mBlock = blockIdx.x * TM;
    const int nBlock = blockIdx.y * TN;

    v8f acc[2][2];
    for (int i = 0; i < 2; ++i)
        for (int j = 0; j < 2; ++j)
            for (int e = 0; e < 8; ++e) acc[i][j][e] = 0.0f;

    const int mfrag = lane & 15;          // M (for A) / N (for B) within 16
    const int kfrag = (lane >> 4) * 2;    // K pair select (lanes 16-31 hold K=2,3)

    for (int k0 = 0; k0 < EMBED; k0 += TK) {
        // ---- stage A tile (TM x TK) row-major into LDS ----
        {
            int col = (tid & 7) * 4;     // 8 threads * 4 floats = 32 = TK
            int row = tid >> 3;          // 32 rows per pass
            for (int p = 0; p < 4; ++p, row += 32) {
                int grow = mBlock + row;
                float4 v = make_float4(0.f, 0.f, 0.f, 0.f);
                if (grow < M)
                    v = *(const float4*)(A + (size_t)grow * EMBED + k0 + col);
                float* dst = &As[row * LDA_S + col];
                dst[0] = v.x; dst[1] = v.y; dst[2] = v.z; dst[3] = v.w;
            }
        }
        // ---- stage B tile (TK x TN) transposed into LDS: Bs[n][k] ----
        {
            int col = (tid & 15) * 4;    // 16 threads * 4 = 64 = TN
            int row = tid >> 4;          // 16 K-rows per pass
            for (int p = 0; p < 2; ++p, row += 16) {
                const float4 v = *(const float4*)(B + (size_t)(k0 + row) * N + nBlock + col);
                Bs[(col + 0) * LDB_S + row] = v.x;
                Bs[(col + 1) * LDB_S + row] = v.y;
                Bs[(col + 2) * LDB_S + row] = v.z;
                Bs[(col + 3) * LDB_S + row] = v.w;
            }
        }
        __syncthreads();

        for (int kk = 0; kk < TK; kk += 4) {
            const int kb = kk + kfrag;
            v2f a0 = *(const v2f*)&As[(wm +      mfrag) * LDA_S + kb];
            v2f a1 = *(const v2f*)&As[(wm + 16 + mfrag) * LDA_S + kb];
            v2f b0 = *(const v2f*)&Bs[(wn +      mfrag) * LDB_S + kb];
            v2f b1 = *(const v2f*)&Bs[(wn + 16 + mfrag) * LDB_S + kb];
            acc[0][0] = __builtin_amdgcn_wmma_f32_16x16x4_f32(false, a0, false, b0, (short)0, acc[0][0], false, false);
            acc[0][1] = __builtin_amdgcn_wmma_f32_16x16x4_f32(false, a0, false, b1, (short)0, acc[0][1], false, false);
            acc[1][0] = __builtin_amdgcn_wmma_f32_16x16x4_f32(false, a1, false, b0, (short)0, acc[1][0], false, false);
            acc[1][1] = __builtin_amdgcn_wmma_f32_16x16x4_f32(false, a1, false, b1, (short)0, acc[1][1], false, false);
        }
        __syncthreads();
    }

    // ---- epilogue: bias (+ residual), C layout: VGPR r -> row r + 8*(lane>=16) ----
    const int colLane = lane & 15;
    const int rowSel  = (lane >> 4) * 8;
    for (int mb = 0; mb < 2; ++mb) {
        for (int nb = 0; nb < 2; ++nb) {
            const int gcol = nBlock + wn + nb * 16 + colLane;
            const float bv = bias[gcol];
            for (int r = 0; r < 8; ++r) {
                const int grow = mBlock + wm + mb * 16 + rowSel + r;
                if (grow < M) {
                    float v = acc[mb][nb][r] + bv;
                    if (residual) v += residual[(size_t)grow * N + gcol];
                    C[(size_t)grow * N + gcol] = v;
                }
            }
        }
    }
}

// ---------------- softmax + bilinear sampling -------------------------------
// One wave per (b, q, head); lane == channel d (DIM == 32 == wave32).
__global__ __launch_bounds__(256)
void msda_sample(const float* __restrict__ value,
                 const float* __restrict__ offL,
                 const float* __restrict__ attnL,
                 const float* __restrict__ refp,
                 float* __restrict__ accOut)
{
    const int lane = threadIdx.x & 31;
    const int gw   = (blockIdx.x * blockDim.x + threadIdx.x) >> 5;
    const int total = MTOT * HEADS;
    if (gw >= total) return;
    const int h   = gw % HEADS;
    const int row = gw / HEADS;          // b*NQ + q
    const int b   = row / NQ;

    const int HW[LEVELS]     = {100, 50, 25, 13};
    const int LSTART[LEVELS] = {0, 10000, 12500, 13125};

    // softmax over 16 attention logits (uniform across lanes, broadcast loads)
    const float* al = attnL + (size_t)row * (HEADS * LEVELS * POINTS) + h * 16;
    float lg[16];
    float mx = al[0];
    for (int i = 0; i < 16; ++i) { lg[i] = al[i]; }
    for (int i = 1; i < 16; ++i) mx = fmaxf(mx, lg[i]);
    float s = 0.f;
    for (int i = 0; i < 16; ++i) { lg[i] = __expf(lg[i] - mx); s += lg[i]; }
    const float inv = 1.0f / s;

    const float* ofs = offL + (size_t)row * EMBED + h * (LEVELS * POINTS * 2);
    const float* rp  = refp + (size_t)row * (LEVELS * 2);

    float acc = 0.f;
    for (int l = 0; l < LEVELS; ++l) {
        const int   H = HW[l], W = HW[l];
        const float fW = (float)W, fH = (float)H;
        const float rx = rp[l * 2 + 0];
        const float ry = rp[l * 2 + 1];
        const size_t vbase = ((size_t)b * NQ + LSTART[l]) * EMBED + h * DIM + lane;
        for (int p = 0; p < POINTS; ++p) {
            const float ox = ofs[l * (POINTS * 2) + p * 2 + 0];
            const float oy = ofs[l * (POINTS * 2) + p * 2 + 1];
            const float x = (rx + ox / fW) * fW - 0.5f;
            const float y = (ry + oy / fH) * fH - 0.5f;
            const float x0f = floorf(x), y0f = floorf(y);
            const float lx = x - x0f, ly = y - y0f;
            const int x0 = (int)x0f, y0 = (int)y0f;
            const float w = lg[l * POINTS + p] * inv;
            const float w00 = w * (1.f - ly) * (1.f - lx);
            const float w01 = w * (1.f - ly) * lx;
            const float w10 = w * ly * (1.f - lx);
            const float w11 = w * ly * lx;

            const bool yi0 = (y0 >= 0) && (y0 < H);
            const bool yi1 = (y0 + 1 >= 0) && (y0 + 1 < H);
            const bool xi0 = (x0 >= 0) && (x0 < W);
            const bool xi1 = (x0 + 1 >= 0) && (x0 + 1 < W);

            float v00 = 0.f, v01 = 0.f, v10 = 0.f, v11 = 0.f;
            if (yi0 && xi0) v00 = value[vbase + (size_t)(y0 * W + x0) * EMBED];
            if (yi0 && xi1) v01 = value[vbase + (size_t)(y0 * W + x0 + 1) * EMBED];
            if (yi1 && xi0) v10 = value[vbase + (size_t)((y0 + 1) * W + x0) * EMBED];
            if (yi1 && xi1) v11 = value[vbase + (size_t)((y0 + 1) * W + x0 + 1) * EMBED];
            acc += w00 * v00 + w01 * v01 + w10 * v10 + w11 * v11;
        }
    }
    accOut[(size_t)row * EMBED + h * DIM + lane] = acc;
}

// ---------------- host launch ------------------------------------------------
extern "C" void kernel_launch(void* const* d_in, const int* in_sizes, int n_in,
                              void* d_out, int out_size, void* d_ws, size_t ws_size,
                              hipStream_t stream) {
    const float* query  = (const float*)d_in[0];   // (BS, NQ, 256)
    const float* refp   = (const float*)d_in[1];   // (BS, NQ, 4, 2)
    //           d_in[2] = spatial_shapes (compile-time constants)
    const float* W_value = (const float*)d_in[3];
    const float* b_value = (const float*)d_in[4];
    const float* W_off   = (const float*)d_in[5];
    const float* b_off   = (const float*)d_in[6];
    const float* W_attn  = (const float*)d_in[7];
    const float* b_attn  = (const float*)d_in[8];
    const float* W_out   = (const float*)d_in[9];
    const float* b_out   = (const float*)d_in[10];
    float* out = (float*)d_out;

    const int M = MTOT;
    float* ws = (float*)d_ws;
    float* val_ws  = ws;                                   // M*256
    float* offL_ws = val_ws  + (size_t)M * EMBED;          // M*256
    float* attn_ws = offL_ws + (size_t)M * EMBED;          // M*128
    float* samp_ws = attn_ws + (size_t)M * (HEADS * LEVELS * POINTS); // M*256

    dim3 block(256);
    const int mTiles = (M + TM - 1) / TM;

    // 1) value = query @ W_value + b_value
    msda_gemm_wmma_f32<<<dim3(mTiles, EMBED / TN), block, 0, stream>>>(
        query, W_value, b_value, nullptr, val_ws, M, EMBED);
    // 2) off logits = query @ W_off + b_off
    msda_gemm_wmma_f32<<<dim3(mTiles, EMBED / TN), block, 0, stream>>>(
        query, W_off, b_off, nullptr, offL_ws, M, EMBED);
    // 3) attn logits = query @ W_attn + b_attn   (N = 128)
    msda_gemm_wmma_f32<<<dim3(mTiles, 128 / TN), block, 0, stream>>>(
        query, W_attn, b_attn, nullptr, attn_ws, M, 128);

    // 4) softmax + bilinear sampling: one wave per (b,q,head)
    {
        const long long waves = (long long)M * HEADS;           // 212704
        const int blocks = (int)((waves * 32 + 255) / 256);     // 26588
        msda_sample<<<blocks, block, 0, stream>>>(val_ws, offL_ws, attn_ws, refp, samp_ws);
    }

    // 5) out = sampled @ W_out + b_out + query (residual)
    msda_gemm_wmma_f32<<<dim3(mTiles, EMBED / TN), block, 0, stream>>>(
        samp_ws, W_out, b_out, query, out, M, EMBED);
}